// CRnnGenerator_6725918786189
// MI455X (gfx1250) — compile-verified
//
#include <hip/hip_runtime.h>
#include <hip/hip_bf16.h>
#include <math.h>

#define NF      196608
#define TWO_NF  393216
#define HD      128
#define SEQ     64
#define SPLITK  96
#define KC      (NF / SPLITK)   /* 2048 */

typedef __attribute__((ext_vector_type(2))) float v2f;
typedef __attribute__((ext_vector_type(4))) float v4f;
typedef __attribute__((ext_vector_type(8))) float v8f;

__device__ __forceinline__ v8f wmma_f32(v2f a, v2f b, v8f c) {
  return __builtin_amdgcn_wmma_f32_16x16x4_f32(false, a, false, b, (short)0, c, false, false);
}

__device__ __forceinline__ float sigm(float x) { return 1.0f / (1.0f + __expf(-x)); }

// ---------------------------------------------------------------------------
// Az = Wz (128 x NF) @ Z^T (NF x 64), split-K partials. One block per K-slice,
// 8 waves = 8 M-tiles, each wave holds 4 N-tile accumulators (N=64).
// K processed 8 at a time with b128 loads using a K-permutation shared by A
// and B: WMMA#1 takes logical K {k,k+1,k+4,k+5}, WMMA#2 {k+2,k+3,k+6,k+7}.
// ---------------------------------------------------------------------------
__global__ __launch_bounds__(256) void k_az(const float* __restrict__ fc1w,
                                            const float* __restrict__ z,
                                            float* __restrict__ azp) {
  const int ks   = blockIdx.x;
  const int wave = threadIdx.x >> 5;
  const int l    = threadIdx.x & 31;
  const int lm   = l & 15, half = l >> 4;
  const int hd0  = wave * 16;

  const float* arow = fc1w + (size_t)(hd0 + lm) * TWO_NF;      // Wz row
  const float* br0  = z + (size_t)( 0 + lm) * NF;
  const float* br1  = z + (size_t)(16 + lm) * NF;
  const float* br2  = z + (size_t)(32 + lm) * NF;
  const float* br3  = z + (size_t)(48 + lm) * NF;

  v8f a0 = {0,0,0,0,0,0,0,0}, a1 = a0, a2 = a0, a3 = a0;
  const int k0 = ks * KC, k1 = k0 + KC;
  for (int k = k0; k < k1; k += 8) {
    const int o = k + 4 * half;
    v4f a  = *(const v4f*)(arow + o);
    v4f b0 = *(const v4f*)(br0 + o);
    v4f b1 = *(const v4f*)(br1 + o);
    v4f b2 = *(const v4f*)(br2 + o);
    v4f b3 = *(const v4f*)(br3 + o);
    v2f alo = {a.x, a.y}, ahi = {a.z, a.w};
    a0 = wmma_f32(alo, (v2f){b0.x, b0.y}, a0);
    a1 = wmma_f32(alo, (v2f){b1.x, b1.y}, a1);
    a2 = wmma_f32(alo, (v2f){b2.x, b2.y}, a2);
    a3 = wmma_f32(alo, (v2f){b3.x, b3.y}, a3);
    a0 = wmma_f32(ahi, (v2f){b0.z, b0.w}, a0);
    a1 = wmma_f32(ahi, (v2f){b1.z, b1.w}, a1);
    a2 = wmma_f32(ahi, (v2f){b2.z, b2.w}, a2);
    a3 = wmma_f32(ahi, (v2f){b3.z, b3.w}, a3);
  }

  float* dst = azp + (size_t)ks * (HD * SEQ);
#pragma unroll
  for (int v = 0; v < 8; ++v) {
    const int m = hd0 + v + 8 * half;
    dst[m * SEQ +  0 + lm] = a0[v];
    dst[m * SEQ + 16 + lm] = a1[v];
    dst[m * SEQ + 32 + lm] = a2[v];
    dst[m * SEQ + 48 + lm] = a3[v];
  }
}

// ---------------------------------------------------------------------------
// R = Wp (128 x NF) @ [fc2_w (NF x 128) | prev_gen | fc2_b | 0...] -> 128 x 144
// cols 0..127 = M, col 128 = Wp@prev_gen, col 129 = Wp@fc2_b. Split-K partials.
// A via b128 (permuted-K); B is transposed-strided in fc2_w, so lane-coalesced
// scalar loads from 4 consecutive rows per 8-K step (shared across all waves).
// ---------------------------------------------------------------------------
__global__ __launch_bounds__(256) void k_m(const float* __restrict__ fc1w,
                                           const float* __restrict__ fc2w,
                                           const float* __restrict__ prevg,
                                           const float* __restrict__ fc2b,
                                           float* __restrict__ mp) {
  const int ks   = blockIdx.x;
  const int wave = threadIdx.x >> 5;
  const int l    = threadIdx.x & 31;
  const int lm   = l & 15, half = l >> 4;
  const int hd0  = wave * 16;

  const float* arow = fc1w + (size_t)(hd0 + lm) * TWO_NF + NF;  // Wp row

  v8f acc[9];
#pragma unroll
  for (int i = 0; i < 9; ++i) acc[i] = (v8f){0,0,0,0,0,0,0,0};

  const int k0 = ks * KC, k1 = k0 + KC;
  for (int k = k0; k < k1; k += 8) {
    const int o = k + 4 * half;
    v4f a = *(const v4f*)(arow + o);
    v2f alo = {a.x, a.y}, ahi = {a.z, a.w};
    const float* brow = fc2w + (size_t)o * HD;   // rows o..o+3, cols contiguous
#pragma unroll
    for (int nt = 0; nt < 8; ++nt) {
      const int cl = nt * 16 + lm;
      v2f blo = { brow[cl],          brow[HD + cl] };
      v2f bhi = { brow[2 * HD + cl], brow[3 * HD + cl] };
      acc[nt] = wmma_f32(alo, blo, acc[nt]);
      acc[nt] = wmma_f32(ahi, bhi, acc[nt]);
    }
    // extra tile: col 0 = prev_gen, col 1 = fc2_b, cols 2..15 = 0
    v4f pv  = *(const v4f*)(prevg + o);
    v4f bv  = *(const v4f*)(fc2b + o);
    v4f sel = (lm == 0) ? pv : ((lm == 1) ? bv : (v4f){0.0f, 0.0f, 0.0f, 0.0f});
    acc[8] = wmma_f32(alo, (v2f){sel.x, sel.y}, acc[8]);
    acc[8] = wmma_f32(ahi, (v2f){sel.z, sel.w}, acc[8]);
  }

  float* dst = mp + (size_t)ks * (HD * 144);
#pragma unroll
  for (int v = 0; v < 8; ++v) {
    const int m = hd0 + v + 8 * half;
#pragma unroll
    for (int nt = 0; nt < 9; ++nt)
      dst[m * 144 + nt * 16 + lm] = acc[nt][v];
  }
}

// ---------------------------------------------------------------------------
// Deterministic split-K slab reduction: dst[i] = sum_s src[s*stride + i]
// ---------------------------------------------------------------------------
__global__ void k_reduce(const float* __restrict__ src, float* __restrict__ dst,
                         int n, int slabs, int stride) {
  const int i = blockIdx.x * blockDim.x + threadIdx.x;
  if (i < n) {
    float s = 0.0f;
    for (int j = 0; j < slabs; ++j) s += src[(size_t)j * stride + i];
    dst[i] = s;
  }
}

// ---------------------------------------------------------------------------
// Serial 64-step recurrence in 128-dim space. One block, 512 threads.
// xin_t = relu(Az[:,t] + (t==0 ? R[:,128] : M@h2 + R[:,129]) + fc1_b)
// two LSTM cells; records h2 per step; writes final states to d_out tail.
// ---------------------------------------------------------------------------
__global__ __launch_bounds__(512) void k_rec(
    const float* __restrict__ Az, const float* __restrict__ R,
    const float* __restrict__ fc1b,
    const float* __restrict__ wih1, const float* __restrict__ whh1,
    const float* __restrict__ bih1, const float* __restrict__ bhh1,
    const float* __restrict__ wih2, const float* __restrict__ whh2,
    const float* __restrict__ bih2, const float* __restrict__ bhh2,
    const float* __restrict__ h1i, const float* __restrict__ c1i,
    const float* __restrict__ h2i, const float* __restrict__ c2i,
    float* __restrict__ h2seq, float* __restrict__ st_out) {
  __shared__ float h1[HD], c1[HD], h2[HD], c2[HD], xin[HD], g[4 * HD];
  const int tid = threadIdx.x;
  if (tid < HD) { h1[tid] = h1i[tid]; c1[tid] = c1i[tid]; h2[tid] = h2i[tid]; c2[tid] = c2i[tid]; }
  __syncthreads();

  for (int t = 0; t < SEQ; ++t) {
    if (tid < HD) {
      float u;
      if (t == 0) {
        u = R[tid * 144 + 128];                    // Wp @ prev_gen
      } else {
        u = R[tid * 144 + 129];                    // Wp @ fc2_b
        const float* Mr = R + tid * 144;
#pragma unroll 8
        for (int j = 0; j < HD; ++j) u = fmaf(Mr[j], h2[j], u);
      }
      float x = Az[tid * SEQ + t] + u + fc1b[tid];
      xin[tid] = fmaxf(x, 0.0f);
    }
    __syncthreads();
    {
      const float* wi = wih1 + tid * HD;
      const float* wh = whh1 + tid * HD;
      float s = bih1[tid] + bhh1[tid];
#pragma unroll 8
      for (int j = 0; j < HD; ++j) s = fmaf(wi[j], xin[j], fmaf(wh[j], h1[j], s));
      g[tid] = s;
    }
    __syncthreads();
    if (tid < HD) {                                 // gates ordered i,f,g,o
      float ig = sigm(g[tid]),           fg = sigm(g[HD + tid]);
      float gg = tanhf(g[2 * HD + tid]), og = sigm(g[3 * HD + tid]);
      float cn = fg * c1[tid] + ig * gg;
      c1[tid] = cn;
      h1[tid] = og * tanhf(cn);
    }
    __syncthreads();
    {
      const float* wi = wih2 + tid * HD;
      const float* wh = whh2 + tid * HD;
      float s = bih2[tid] + bhh2[tid];
#pragma unroll 8
      for (int j = 0; j < HD; ++j) s = fmaf(wi[j], h1[j], fmaf(wh[j], h2[j], s));
      g[tid] = s;
    }
    __syncthreads();
    if (tid < HD) {
      float ig = sigm(g[tid]),           fg = sigm(g[HD + tid]);
      float gg = tanhf(g[2 * HD + tid]), og = sigm(g[3 * HD + tid]);
      float cn = fg * c2[tid] + ig * gg;
      c2[tid] = cn;
      float hn = og * tanhf(cn);
      h2[tid] = hn;
      h2seq[t * HD + tid] = hn;
    }
    __syncthreads();
  }

  if (tid < HD) {
    st_out[tid]          = h1[tid];
    st_out[HD + tid]     = c1[tid];
    st_out[2 * HD + tid] = h2[tid];
    st_out[3 * HD + tid] = c2[tid];
  }
}

// ---------------------------------------------------------------------------
// out[t, feat] = fc2_w[feat,:] @ h2_t + fc2_b[feat] + y[t, feat]
// GEMM (NF x 128) @ (128 x 64). wave = one 16-feature tile, 4 t-tile accs.
// b128 permuted-K loads.
// ---------------------------------------------------------------------------
__global__ __launch_bounds__(256) void k_out(const float* __restrict__ fc2w,
                                             const float* __restrict__ h2seq,
                                             const float* __restrict__ fc2b,
                                             const float* __restrict__ y,
                                             float* __restrict__ out) {
  const int wave = threadIdx.x >> 5;
  const int l    = threadIdx.x & 31;
  const int lm   = l & 15, half = l >> 4;
  const int m0   = blockIdx.x * 128 + wave * 16;

  const float* arow = fc2w  + (size_t)(m0 + lm) * HD;
  const float* br0  = h2seq + (size_t)( 0 + lm) * HD;
  const float* br1  = h2seq + (size_t)(16 + lm) * HD;
  const float* br2  = h2seq + (size_t)(32 + lm) * HD;
  const float* br3  = h2seq + (size_t)(48 + lm) * HD;

  v8f a0 = {0,0,0,0,0,0,0,0}, a1 = a0, a2 = a0, a3 = a0;
#pragma unroll 4
  for (int k = 0; k < HD; k += 8) {
    const int o = k + 4 * half;
    v4f a  = *(const v4f*)(arow + o);
    v4f b0 = *(const v4f*)(br0 + o);
    v4f b1 = *(const v4f*)(br1 + o);
    v4f b2 = *(const v4f*)(br2 + o);
    v4f b3 = *(const v4f*)(br3 + o);
    v2f alo = {a.x, a.y}, ahi = {a.z, a.w};
    a0 = wmma_f32(alo, (v2f){b0.x, b0.y}, a0);
    a1 = wmma_f32(alo, (v2f){b1.x, b1.y}, a1);
    a2 = wmma_f32(alo, (v2f){b2.x, b2.y}, a2);
    a3 = wmma_f32(alo, (v2f){b3.x, b3.y}, a3);
    a0 = wmma_f32(ahi, (v2f){b0.z, b0.w}, a0);
    a1 = wmma_f32(ahi, (v2f){b1.z, b1.w}, a1);
    a2 = wmma_f32(ahi, (v2f){b2.z, b2.w}, a2);
    a3 = wmma_f32(ahi, (v2f){b3.z, b3.w}, a3);
  }

#pragma unroll
  for (int v = 0; v < 8; ++v) {
    const int feat = m0 + v + 8 * half;
    const float bias = fc2b[feat];
    size_t i0 = (size_t)( 0 + lm) * NF + feat; out[i0] = a0[v] + bias + y[i0];
    size_t i1 = (size_t)(16 + lm) * NF + feat; out[i1] = a1[v] + bias + y[i1];
    size_t i2 = (size_t)(32 + lm) * NF + feat; out[i2] = a2[v] + bias + y[i2];
    size_t i3 = (size_t)(48 + lm) * NF + feat; out[i3] = a3[v] + bias + y[i3];
  }
}

// ---------------------------------------------------------------------------
// Workspace layout (floats)
// ---------------------------------------------------------------------------
#define AZP_OFF 0u                                   /* SPLITK*128*64  = 786432 */
#define MP_OFF  (AZP_OFF + SPLITK * HD * SEQ)        /* SPLITK*128*144 = 1769472 */
#define AZ_OFF  (MP_OFF + SPLITK * HD * 144)         /* 8192 */
#define R_OFF   (AZ_OFF + HD * SEQ)                  /* 18432 */
#define H2_OFF  (R_OFF + HD * 144)                   /* 8192 */

extern "C" void kernel_launch(void* const* d_in, const int* in_sizes, int n_in,
                              void* d_out, int out_size, void* d_ws, size_t ws_size,
                              hipStream_t stream) {
  (void)in_sizes; (void)n_in; (void)out_size; (void)ws_size;

  const float* z    = (const float*)d_in[0];
  const float* y    = (const float*)d_in[1];
  const float* pg   = (const float*)d_in[2];
  const float* h1i  = (const float*)d_in[3];
  const float* c1i  = (const float*)d_in[4];
  const float* h2i  = (const float*)d_in[5];
  const float* c2i  = (const float*)d_in[6];
  const float* fc1w = (const float*)d_in[7];
  const float* fc1b = (const float*)d_in[8];
  const float* wih1 = (const float*)d_in[9];
  const float* whh1 = (const float*)d_in[10];
  const float* bih1 = (const float*)d_in[11];
  const float* bhh1 = (const float*)d_in[12];
  const float* wih2 = (const float*)d_in[13];
  const float* whh2 = (const float*)d_in[14];
  const float* bih2 = (const float*)d_in[15];
  const float* bhh2 = (const float*)d_in[16];
  const float* fc2w = (const float*)d_in[17];
  const float* fc2b = (const float*)d_in[18];

  float* out = (float*)d_out;
  float* ws  = (float*)d_ws;
  float* azp = ws + AZP_OFF;
  float* mp  = ws + MP_OFF;
  float* az  = ws + AZ_OFF;
  float* r   = ws + R_OFF;
  float* h2s = ws + H2_OFF;

  k_az<<<dim3(SPLITK), 256, 0, stream>>>(fc1w, z, azp);
  k_m<<<dim3(SPLITK), 256, 0, stream>>>(fc1w, fc2w, pg, fc2b, mp);
  k_reduce<<<dim3((HD * SEQ + 255) / 256), 256, 0, stream>>>(azp, az, HD * SEQ, SPLITK, HD * SEQ);
  k_reduce<<<dim3((HD * 144 + 255) / 256), 256, 0, stream>>>(mp, r, HD * 144, SPLITK, HD * 144);
  k_rec<<<dim3(1), 512, 0, stream>>>(az, r, fc1b,
                                     wih1, whh1, bih1, bhh1,
                                     wih2, whh2, bih2, bhh2,
                                     h1i, c1i, h2i, c2i,
                                     h2s, out + (size_t)SEQ * NF);
  k_out<<<dim3(NF / 128), 256, 0, stream>>>(fc2w, h2s, fc2b, y, out);
}